// convTransBlock_9165460210068
// MI455X (gfx1250) — compile-verified
//
#include <hip/hip_runtime.h>
#include <hip/hip_bf16.h>

typedef float v2f __attribute__((ext_vector_type(2)));
typedef float v8f __attribute__((ext_vector_type(8)));

constexpr int CC = 64;        // channels
constexpr int HH = 256;
constexpr int WWp = 256;
constexpr int NP = HH * WWp;  // 65536 pixels

__device__ inline v8f wmma_f32_4(v2f a, v2f b, v8f c) {
  // D = A(16x4 f32) * B(4x16 f32) + C(16x16 f32)
  return __builtin_amdgcn_wmma_f32_16x16x4_f32(
      /*neg_a=*/false, a, /*neg_b=*/false, b,
      /*c_mod=*/(short)0, c, /*reuse_a=*/false, /*reuse_b=*/false);
}

__device__ inline float gelu_f(float v) {
  return 0.5f * v * (1.0f + erff(v * 0.70710678118654752f));
}

// Stage a 64ch x 64px f32 tile (global row stride NP) into LDS laid out as
// [64][68] floats, using gfx1250 async VMEM->LDS copies (no VGPR roundtrip).
// Each of 256 threads issues 4 x b128 transfers; per-wave s_wait_asynccnt.
__device__ inline void stage_tile_async(const float* __restrict__ src,
                                        uint32_t lds_base, int tid) {
  #pragma unroll
  for (int i = 0; i < 4; ++i) {
    const int idx = i * 256 + tid;       // 0..1023 chunks of 16B
    const int c = idx >> 4;              // channel row 0..63
    const int col = (idx & 15) << 2;     // float column 0..60 step 4
    const uint32_t loff = lds_base + (uint32_t)((c * 68 + col) * 4);
    const float* g = src + (size_t)c * NP + col;
    asm volatile("global_load_async_to_lds_b128 %0, %1, off"
                 :: "v"(loff), "v"(g)
                 : "memory");
  }
  asm volatile("s_wait_asynccnt 0" ::: "memory");
}

// ---------------------------------------------------------------------------
// K1: qkv = 1x1 conv (GEMM [192 x 64] x [64 x N]) + bias; relu on q,k rows.
// Block: 256 threads (8 waves), 64-pixel tile staged async into LDS.
// ---------------------------------------------------------------------------
__global__ __launch_bounds__(256) void qkv_kernel(
    const float* __restrict__ x, const float* __restrict__ w,
    const float* __restrict__ bias, float* __restrict__ qkv) {
  __shared__ float sx[64][68];
  const int b = blockIdx.y;
  const int n0 = blockIdx.x * 64;
  const int tid = threadIdx.x;
  const float* xb = x + (size_t)b * CC * NP;
  const uint32_t lds_base = (uint32_t)(uintptr_t)&sx[0][0];
  stage_tile_async(xb + n0, lds_base, tid);
  __syncthreads();
  const int wave = tid >> 5, lane = tid & 31;
  const int lhi = lane >> 4, llo = lane & 15;
  float* outb = qkv + (size_t)b * 3 * CC * NP;
  for (int t = wave; t < 48; t += 8) {
    const int oc0 = (t >> 2) * 16, px0 = (t & 3) * 16;
    v8f acc;
    #pragma unroll
    for (int r = 0; r < 8; ++r) acc[r] = bias[oc0 + r + lhi * 8];
    #pragma unroll 4
    for (int k0 = 0; k0 < 64; k0 += 4) {
      const int ka = k0 + (lhi << 1);
      v2f a, bb;
      a.x = w[(oc0 + llo) * 64 + ka];
      a.y = w[(oc0 + llo) * 64 + ka + 1];
      bb.x = sx[ka][px0 + llo];
      bb.y = sx[ka + 1][px0 + llo];
      acc = wmma_f32_4(a, bb, acc);
    }
    #pragma unroll
    for (int r = 0; r < 8; ++r) {
      const int rr = oc0 + r + lhi * 8;
      float val = acc[r];
      if (rr < 128) val = fmaxf(val, 0.0f);  // relu on q,k planes
      outb[(size_t)rr * NP + n0 + px0 + llo] = val;
    }
  }
}

// ---------------------------------------------------------------------------
// K2a: zero the kv/ksum accumulators.
// ---------------------------------------------------------------------------
__global__ void zero_kernel(float* __restrict__ p, int n) {
  int i = blockIdx.x * 256 + threadIdx.x;
  if (i < n) p[i] = 0.0f;
}

// ---------------------------------------------------------------------------
// K2: kv[b,h,d,e] = sum_n kU[d,n]*v[e,n]; ksum[b,h,d] = sum_n kU[d,n].
// kU built on the fly from 3x3 unfold of k (LDS halo rows). 288 threads,
// each owns 2 (d,e) pairs of the 72x8 accumulator; fp32 atomics at the end.
// ---------------------------------------------------------------------------
__global__ __launch_bounds__(288) void kv_kernel(
    const float* __restrict__ qkv, float* __restrict__ kv,
    float* __restrict__ ksum) {
  __shared__ float sk[8][3][WWp + 2];
  __shared__ float sv[8][WWp];
  const int bh = blockIdx.x;  // 0..15
  const int b = bh >> 3, h = bh & 7;
  const int yc = blockIdx.y;  // 16 row-chunks of 16 rows
  const int tid = threadIdx.x;
  const float* kbase = qkv + ((size_t)b * 3 + 1) * CC * NP + (size_t)h * 8 * NP;
  const float* vbase = qkv + ((size_t)b * 3 + 2) * CC * NP + (size_t)h * 8 * NP;
  const int d0 = tid >> 3, e0 = tid & 7;  // pairs (d0,e0) and (d0+36,e0)
  const int ci0 = d0 / 9, pp0 = d0 % 9;
  const int d1 = d0 + 36;
  const int ci1 = d1 / 9, pp1 = d1 % 9;
  const int i0 = pp0 / 3, j0 = pp0 % 3;
  const int i1 = pp1 / 3, j1 = pp1 % 3;
  float acc0 = 0.0f, acc1 = 0.0f, accs0 = 0.0f, accs1 = 0.0f;
  for (int yy = 0; yy < 16; ++yy) {
    const int y = yc * 16 + yy;
    __syncthreads();
    for (int idx = tid; idx < 8 * 3 * (WWp + 2); idx += 288) {
      int ch = idx / (3 * (WWp + 2));
      int rem = idx % (3 * (WWp + 2));
      int ry = rem / (WWp + 2), rx = rem % (WWp + 2);
      int gy = y + ry - 1, gx = rx - 1;
      float v = 0.0f;
      if (gy >= 0 && gy < HH && gx >= 0 && gx < WWp)
        v = kbase[(size_t)ch * NP + gy * WWp + gx];
      sk[ch][ry][rx] = v;
    }
    for (int idx = tid; idx < 8 * WWp; idx += 288) {
      int ch = idx >> 8, px = idx & 255;
      sv[ch][px] = vbase[(size_t)ch * NP + y * WWp + px];
    }
    __syncthreads();
    for (int xx = 0; xx < WWp; ++xx) {
      const float k0v = sk[ci0][i0][xx + j0];
      const float k1v = sk[ci1][i1][xx + j1];
      const float vv = sv[e0][xx];
      acc0 = fmaf(k0v, vv, acc0);
      acc1 = fmaf(k1v, vv, acc1);
      if (e0 == 0) { accs0 += k0v; accs1 += k1v; }
    }
  }
  float* kvb = kv + (size_t)bh * 72 * 8;
  atomicAdd(&kvb[d0 * 8 + e0], acc0);
  atomicAdd(&kvb[d1 * 8 + e0], acc1);
  if (e0 == 0) {
    float* ksb = ksum + (size_t)bh * 72;
    atomicAdd(&ksb[d0], accs0);
    atomicAdd(&ksb[d1], accs1);
  }
}

// ---------------------------------------------------------------------------
// K3: o[b,8h+e,n] = (sum_d qU[d,n]*kv[d,e]) / (sum_d qU[d,n]*ksum[d] + eps)
// One block per (row, b*h); 256 threads = 256 pixel columns.
// ---------------------------------------------------------------------------
__global__ __launch_bounds__(256) void attnout_kernel(
    const float* __restrict__ qkv, const float* __restrict__ kv,
    const float* __restrict__ ksum, float* __restrict__ o) {
  __shared__ float sq[8][3][WWp + 2];
  __shared__ float skv[72][8];
  __shared__ float sks[72];
  const int bh = blockIdx.y;
  const int b = bh >> 3, h = bh & 7;
  const int y = blockIdx.x;
  const int tid = threadIdx.x;
  const float* qbase = qkv + (size_t)b * 3 * CC * NP + (size_t)h * 8 * NP;
  for (int idx = tid; idx < 576; idx += 256)
    skv[idx >> 3][idx & 7] = kv[(size_t)bh * 576 + idx];
  if (tid < 72) sks[tid] = ksum[(size_t)bh * 72 + tid];
  for (int idx = tid; idx < 8 * 3 * (WWp + 2); idx += 256) {
    int ch = idx / (3 * (WWp + 2));
    int rem = idx % (3 * (WWp + 2));
    int ry = rem / (WWp + 2), rx = rem % (WWp + 2);
    int gy = y + ry - 1, gx = rx - 1;
    float v = 0.0f;
    if (gy >= 0 && gy < HH && gx >= 0 && gx < WWp)
      v = qbase[(size_t)ch * NP + gy * WWp + gx];
    sq[ch][ry][rx] = v;
  }
  __syncthreads();
  float acc[8] = {0, 0, 0, 0, 0, 0, 0, 0};
  float den = 0.0f;
  #pragma unroll
  for (int ci = 0; ci < 8; ++ci) {
    #pragma unroll
    for (int pp = 0; pp < 9; ++pp) {
      const int d = ci * 9 + pp;
      const float qv = sq[ci][pp / 3][tid + pp % 3];
      den = fmaf(qv, sks[d], den);
      #pragma unroll
      for (int e = 0; e < 8; ++e) acc[e] = fmaf(qv, skv[d][e], acc[e]);
    }
  }
  const float inv = 1.0f / (den + 1e-6f);
  float* ob = o + (size_t)b * CC * NP + (size_t)h * 8 * NP + y * WWp + tid;
  #pragma unroll
  for (int e = 0; e < 8; ++e) ob[(size_t)e * NP] = acc[e] * inv;
}

// ---------------------------------------------------------------------------
// K4: x1 = x + proj3x3(o) + bias. Implicit GEMM restructured tap-major:
// out[co,px] = sum_{pp=0..8} sum_{ci=0..63} w[co][ci][pp] * so[ci][ry][px+dx]
// -> 9 x (K=64 WMMA-f32 GEMM) with affine addressing (no div/mod in loop).
// ---------------------------------------------------------------------------
__global__ __launch_bounds__(256) void proj_kernel(
    const float* __restrict__ o, const float* __restrict__ w,
    const float* __restrict__ bias, const float* __restrict__ x,
    float* __restrict__ x1) {
  __shared__ float so[64][3][66];
  const int b = blockIdx.z;
  const int y = blockIdx.y;
  const int n0 = blockIdx.x * 64;
  const int tid = threadIdx.x;
  const float* ob = o + (size_t)b * CC * NP;
  for (int idx = tid; idx < 64 * 3 * 66; idx += 256) {
    int ch = idx / (3 * 66);
    int rem = idx % (3 * 66);
    int ry = rem / 66, rx = rem % 66;
    int gy = y + ry - 1, gx = n0 + rx - 1;
    float v = 0.0f;
    if (gy >= 0 && gy < HH && gx >= 0 && gx < WWp)
      v = ob[(size_t)ch * NP + gy * WWp + gx];
    so[ch][ry][rx] = v;
  }
  __syncthreads();
  const int wave = tid >> 5, lane = tid & 31;
  const int lhi = lane >> 4, llo = lane & 15;
  const float* xb = x + (size_t)b * CC * NP;
  float* x1b = x1 + (size_t)b * CC * NP;
  for (int t = wave; t < 16; t += 8) {
    const int co0 = (t >> 2) * 16, px0 = (t & 3) * 16;
    v8f acc;
    #pragma unroll
    for (int r = 0; r < 8; ++r) acc[r] = bias[co0 + r + lhi * 8];
    for (int pp = 0; pp < 9; ++pp) {
      const int ry = pp / 3, dx = pp % 3;          // once per 16 k-steps
      const float* wrow = w + (co0 + llo) * 576 + pp;   // stride 9 over ci
      const float* brow = &so[0][ry][px0 + llo + dx];   // stride 3*66 over ci
      #pragma unroll 4
      for (int k0 = 0; k0 < 64; k0 += 4) {
        const int ka = k0 + (lhi << 1);
        v2f a, bb;
        a.x = wrow[ka * 9];
        a.y = wrow[ka * 9 + 9];
        bb.x = brow[ka * (3 * 66)];
        bb.y = brow[ka * (3 * 66) + (3 * 66)];
        acc = wmma_f32_4(a, bb, acc);
      }
    }
    #pragma unroll
    for (int r = 0; r < 8; ++r) {
      const int rr = co0 + r + lhi * 8;
      const size_t off = (size_t)rr * NP + (size_t)y * WWp + n0 + px0 + llo;
      x1b[off] = xb[off] + acc[r];
    }
  }
}

// ---------------------------------------------------------------------------
// K5/K7: 1x1 conv GEMM [64x64] x [64xN] + bias, optional gelu / residual.
// ---------------------------------------------------------------------------
__global__ __launch_bounds__(256) void conv1x1_kernel(
    const float* __restrict__ in, const float* __restrict__ w,
    const float* __restrict__ bias, const float* __restrict__ res,
    float* __restrict__ out, int apply_gelu) {
  __shared__ float sx[64][68];
  const int b = blockIdx.y;
  const int n0 = blockIdx.x * 64;
  const int tid = threadIdx.x;
  const float* inb = in + (size_t)b * CC * NP;
  const uint32_t lds_base = (uint32_t)(uintptr_t)&sx[0][0];
  stage_tile_async(inb + n0, lds_base, tid);
  __syncthreads();
  const int wave = tid >> 5, lane = tid & 31;
  const int lhi = lane >> 4, llo = lane & 15;
  float* outb = out + (size_t)b * CC * NP;
  const float* resb = res ? res + (size_t)b * CC * NP : nullptr;
  for (int t = wave; t < 16; t += 8) {
    const int oc0 = (t >> 2) * 16, px0 = (t & 3) * 16;
    v8f acc;
    #pragma unroll
    for (int r = 0; r < 8; ++r) acc[r] = bias[oc0 + r + lhi * 8];
    #pragma unroll 4
    for (int k0 = 0; k0 < 64; k0 += 4) {
      const int ka = k0 + (lhi << 1);
      v2f a, bb;
      a.x = w[(oc0 + llo) * 64 + ka];
      a.y = w[(oc0 + llo) * 64 + ka + 1];
      bb.x = sx[ka][px0 + llo];
      bb.y = sx[ka + 1][px0 + llo];
      acc = wmma_f32_4(a, bb, acc);
    }
    #pragma unroll
    for (int r = 0; r < 8; ++r) {
      const int rr = oc0 + r + lhi * 8;
      const size_t off = (size_t)rr * NP + n0 + px0 + llo;
      float val = acc[r];
      if (apply_gelu) val = gelu_f(val);
      if (resb) val += resb[off];
      outb[off] = val;
    }
  }
}

// ---------------------------------------------------------------------------
// K6: depthwise 3x3 + bias + gelu (memory-bound, VALU).
// ---------------------------------------------------------------------------
__global__ __launch_bounds__(256) void dw_kernel(
    const float* __restrict__ h1, const float* __restrict__ w,
    const float* __restrict__ bias, float* __restrict__ h2) {
  const int bc = blockIdx.y;  // b*64 + c
  const int y = blockIdx.x;
  const int tid = threadIdx.x;
  const int ch = bc & 63;
  const float* src = h1 + (size_t)bc * NP;
  float acc = bias[ch];
  #pragma unroll
  for (int i = 0; i < 3; ++i) {
    #pragma unroll
    for (int j = 0; j < 3; ++j) {
      const int gy = y + i - 1, gx = tid + j - 1;
      if (gy >= 0 && gy < HH && gx >= 0 && gx < WWp)
        acc = fmaf(w[ch * 9 + i * 3 + j], src[gy * WWp + gx], acc);
    }
  }
  h2[(size_t)bc * NP + (size_t)y * WWp + tid] = gelu_f(acc);
}

// ---------------------------------------------------------------------------
extern "C" void kernel_launch(void* const* d_in, const int* in_sizes, int n_in,
                              void* d_out, int out_size, void* d_ws,
                              size_t ws_size, hipStream_t stream) {
  const float* x      = (const float*)d_in[0];
  const float* qkv_w  = (const float*)d_in[1];
  const float* qkv_b  = (const float*)d_in[2];
  const float* proj_w = (const float*)d_in[3];
  const float* proj_b = (const float*)d_in[4];
  const float* ffn1_w = (const float*)d_in[5];
  const float* ffn1_b = (const float*)d_in[6];
  const float* dw_w   = (const float*)d_in[7];
  const float* dw_b   = (const float*)d_in[8];
  const float* ffn2_w = (const float*)d_in[9];
  const float* ffn2_b = (const float*)d_in[10];
  float* out = (float*)d_out;

  const size_t plane = (size_t)2 * CC * NP;  // one [2,64,N] tensor
  float* ws = (float*)d_ws;
  float* qkvbuf = ws;                  // [2,192,N]
  float* obuf   = qkvbuf + 3 * plane;  // [2,64,N]
  float* x1buf  = obuf + plane;        // [2,64,N]
  float* kvbuf  = x1buf + plane;       // 16*576
  float* ksbuf  = kvbuf + 16 * 576;    // 16*72
  float* h1buf  = qkvbuf;              // reuse qkv space after attention
  float* h2buf  = qkvbuf + plane;

  qkv_kernel<<<dim3(NP / 64, 2), 256, 0, stream>>>(x, qkv_w, qkv_b, qkvbuf);
  zero_kernel<<<(16 * (576 + 72) + 255) / 256, 256, 0, stream>>>(
      kvbuf, 16 * (576 + 72));
  kv_kernel<<<dim3(16, 16), 288, 0, stream>>>(qkvbuf, kvbuf, ksbuf);
  attnout_kernel<<<dim3(HH, 16), 256, 0, stream>>>(qkvbuf, kvbuf, ksbuf, obuf);
  proj_kernel<<<dim3(WWp / 64, HH, 2), 256, 0, stream>>>(obuf, proj_w, proj_b,
                                                         x, x1buf);
  conv1x1_kernel<<<dim3(NP / 64, 2), 256, 0, stream>>>(x1buf, ffn1_w, ffn1_b,
                                                       nullptr, h1buf, 1);
  dw_kernel<<<dim3(HH, 2 * CC), 256, 0, stream>>>(h1buf, dw_w, dw_b, h2buf);
  conv1x1_kernel<<<dim3(NP / 64, 2), 256, 0, stream>>>(h2buf, ffn2_w, ffn2_b,
                                                       x1buf, out, 0);
}